// DepthAlignment_91096256348296
// MI455X (gfx1250) — compile-verified
//
#include <hip/hip_runtime.h>

// ---------------------------------------------------------------------------
// DepthAlignment on gfx1250 (MI455X)
//  - rigid transform done via V_WMMA_F32_16X16X4_F32 (D = [R^T|t] * [x;y;z;1])
//  - depth streamed global -> LDS via CDNA5 async copy (ASYNCcnt path)
//  - z-buffer min-splat via global_atomic_min_u32 on float bit patterns
// ---------------------------------------------------------------------------

typedef __attribute__((ext_vector_type(2))) float v2f;
typedef __attribute__((ext_vector_type(8))) float v8f;

#define DA_W 3840
#define DA_H 2160
#define DA_N (DA_W * DA_H)          // 8294400 = 32400 * 256
#define FILL_BITS 0x461C4000u       // __float_as_uint(10000.0f)

#define SCALE_F 0.001f
#define D_CX 1919.5f
#define D_CY 1079.5f
#define D_FX 3000.0f
#define D_FY 3000.0f
#define R_CX 1919.5f
#define R_CY 1079.5f
#define R_FX 3050.0f
#define R_FY 3050.0f

// ---------------------------------------------------------------------------
__global__ __launch_bounds__(256) void da_init(uint4* __restrict__ buf) {
    int i = blockIdx.x * 256 + threadIdx.x;       // N/4 elements, exact grid
    buf[i] = make_uint4(FILL_BITS, FILL_BITS, FILL_BITS, FILL_BITS);
}

// ---------------------------------------------------------------------------
__global__ __launch_bounds__(256) void da_splat(const float* __restrict__ depth,
                                                const float* __restrict__ Rm,
                                                const float* __restrict__ tv,
                                                unsigned* __restrict__ zbuf) {
    __shared__ float s_depth[256];
    const int tid  = threadIdx.x;
    const int gid  = blockIdx.x * 256 + tid;      // pixel id, exact grid
    const int lane = tid & 31;

    float d;
#if defined(__gfx1250__)
    {
        // CDNA5 async copy: global -> LDS, tracked by ASYNCcnt
        auto lptr = (__attribute__((address_space(3))) int*)(&s_depth[tid]);
        auto gptr = (__attribute__((address_space(1))) int*)(depth + gid);
#if __has_builtin(__builtin_amdgcn_global_load_async_to_lds_b32)
        __builtin_amdgcn_global_load_async_to_lds_b32(gptr, lptr, 0, 0);
#else
        asm volatile("global_load_async_to_lds_b32 %0, %1, off"
                     :: "v"(lptr), "v"(gptr) : "memory");
#endif
#if __has_builtin(__builtin_amdgcn_s_wait_asynccnt)
        __builtin_amdgcn_s_wait_asynccnt(0);
#else
        asm volatile("s_wait_asynccnt 0" ::: "memory");
#endif
        d = s_depth[tid];                         // ds_load_b32
    }
#else
    d = depth[gid];
#endif

    // unproject: ((u-cx)/fx, (v-cy)/fy, 1) * SCALE * depth
    unsigned upix = (unsigned)gid;
    unsigned u = upix % (unsigned)DA_W;
    unsigned v = upix / (unsigned)DA_W;
    float gx = (((float)u - D_CX) / D_FX) * SCALE_F;
    float gy = (((float)v - D_CY) / D_FY) * SCALE_F;
    float x = d * gx;
    float y = d * gy;
    float z = d * SCALE_F;

    // A matrix (16x4, loop/lane-invariant): rows 0..2 = [R^T | t], rest zero.
    // Layout (ISA 7.12.2, 32-bit A 16x4): lane L<16 -> A[L][0],A[L][1];
    // lane L>=16 -> A[L-16][2],A[L-16][3].
    int m = lane & 15;
    v2f A; A.x = 0.0f; A.y = 0.0f;
    if (m < 3) {
        if (lane < 16) { A.x = Rm[0 * 3 + m]; A.y = Rm[1 * 3 + m]; }
        else           { A.x = Rm[2 * 3 + m]; A.y = tv[m];        }
    }

    // Exchange half-waves so each WMMA sees a full 4x16 B (rows x,y,z,1).
    float xX = __shfl_xor(x, 16, 32);
    float yX = __shfl_xor(y, 16, 32);
    float zX = __shfl_xor(z, 16, 32);

    // B layout (4x16): vgpr0 = row K=0 (lanes 0-15) / K=2 (lanes 16-31),
    //                  vgpr1 = row K=1 (lanes 0-15) / K=3 (lanes 16-31).
    v2f B1, B2;
    if (lane < 16) { B1.x = x;  B1.y = y;    B2.x = xX; B2.y = yX;   }
    else           { B1.x = zX; B1.y = 1.0f; B2.x = z;  B2.y = 1.0f; }

    v8f c0 = {};
    // D rows 0..2 = (X,Y,Z) of the 16 pixels in B's columns; lanes = columns.
    v8f c1 = __builtin_amdgcn_wmma_f32_16x16x4_f32(false, A, false, B1,
                                                   (short)0, c0, false, false);
    v8f c2 = __builtin_amdgcn_wmma_f32_16x16x4_f32(false, A, false, B2,
                                                   (short)0, c0, false, false);

    // c1 lanes 0-15 hold pixels (tile+0..15); c2 lanes 0-15 hold (tile+16..31).
    // Swap c2's results to the upper half so every lane owns its own pixel.
    float X2 = __shfl_xor(c2[0], 16, 32);
    float Y2 = __shfl_xor(c2[1], 16, 32);
    float Z2 = __shfl_xor(c2[2], 16, 32);

    float X = (lane < 16) ? c1[0] : X2;
    float Y = (lane < 16) ? c1[1] : Y2;
    float Z = (lane < 16) ? c1[2] : Z2;

    // perspective projection into RGB camera
    float px, py;
    if (Z == 0.0f) { px = 0.0f; py = 0.0f; }
    else           { px = X / Z * R_FX + R_CX; py = Y / Z * R_FY + R_CY; }
    if (px < 0.0f || px >= (float)DA_W || py < 0.0f || py >= (float)DA_H) {
        px = 0.0f; py = 0.0f;
    }

    // truncation toward zero matches astype(int32)
    int px0 = (int)(px - 0.5f);
    int px1 = (int)(px + 0.5f);
    int py0 = (int)(py - 0.5f);
    int py1 = (int)(py + 0.5f);

    unsigned bits = __float_as_uint(d);  // depth >= 0: uint order == float order

    // after masking: px0,py0 always in range; px1/py1 may equal W/H (dropped)
    atomicMin(&zbuf[py0 * DA_W + px0], bits);
    if (px1 < DA_W)               atomicMin(&zbuf[py0 * DA_W + px1], bits);
    if (py1 < DA_H)               atomicMin(&zbuf[py1 * DA_W + px0], bits);
    if (px1 < DA_W && py1 < DA_H) atomicMin(&zbuf[py1 * DA_W + px1], bits);
}

// ---------------------------------------------------------------------------
__global__ __launch_bounds__(256) void da_finalize(uint4* __restrict__ buf) {
    int i = blockIdx.x * 256 + threadIdx.x;       // N/4 elements, exact grid
    uint4 b = buf[i];
    float4 r;
    r.x = (b.x == FILL_BITS) ? 0.0f : __uint_as_float(b.x);
    r.y = (b.y == FILL_BITS) ? 0.0f : __uint_as_float(b.y);
    r.z = (b.z == FILL_BITS) ? 0.0f : __uint_as_float(b.z);
    r.w = (b.w == FILL_BITS) ? 0.0f : __uint_as_float(b.w);
    reinterpret_cast<float4*>(buf)[i] = r;
}

// ---------------------------------------------------------------------------
extern "C" void kernel_launch(void* const* d_in, const int* in_sizes, int n_in,
                              void* d_out, int out_size, void* d_ws, size_t ws_size,
                              hipStream_t stream) {
    (void)in_sizes; (void)n_in; (void)out_size; (void)d_ws; (void)ws_size;

    const float* depth = (const float*)d_in[0];   // [H,W,1] float32
    const float* Rm    = (const float*)d_in[1];   // [3,3]   float32 row-major
    const float* tv    = (const float*)d_in[2];   // [3]     float32

    unsigned* zbuf = (unsigned*)d_out;            // z-buffer lives in d_out

    // N = 8294400: N/4/256 = 8100 blocks, N/256 = 32400 blocks (exact)
    da_init    <<<8100,  256, 0, stream>>>((uint4*)zbuf);
    da_splat   <<<32400, 256, 0, stream>>>(depth, Rm, tv, zbuf);
    da_finalize<<<8100,  256, 0, stream>>>((uint4*)zbuf);
}